// GCNModule_10788957848201
// MI455X (gfx1250) — compile-verified
//
#include <hip/hip_runtime.h>

typedef float v2f __attribute__((ext_vector_type(2)));
typedef float v8f __attribute__((ext_vector_type(8)));

#define HID 128

// ---------------- degree / normalization ----------------
__global__ void k_deg_init(float* deg, int n) {
    int i = blockIdx.x * blockDim.x + threadIdx.x;
    if (i < n) deg[i] = 1.0f;                      // self-loop
}

__global__ void k_deg_count(const long long* __restrict__ dst, float* deg, int e) {
    int i = blockIdx.x * blockDim.x + threadIdx.x;
    if (i < e) unsafeAtomicAdd(&deg[dst[i]], 1.0f);
}

__global__ void k_deg_rsqrt(float* deg, int n) {
    int i = blockIdx.x * blockDim.x + threadIdx.x;
    if (i < n) deg[i] = rsqrtf(deg[i]);            // deg >= 1 always
}

// ---------------- WMMA GEMM: Y[N,COLS] = X[N,128] @ W[128,COLS] ----------------
// One 16x16 output tile per wave32. A: 16x4 f32 (2 VGPR, K split across lane halves
// per ISA 7.12.2), B: 4x16 f32 (symmetric), C/D: 8 VGPR, M = r + 8*(lane>=16), N = lane&15.
template <int COLS>
__global__ __launch_bounds__((COLS / 16) * 32)
void k_gemm(const float* __restrict__ X, const float* __restrict__ W,
            float* __restrict__ Y) {
    constexpr int NTHREADS = (COLS / 16) * 32;
    constexpr int KCHUNK   = 32;
    __shared__ float sX[16 * HID];                 // 8 KB
    __shared__ float sW[KCHUNK * COLS];            // 16 KB (128) / 8 KB (64)

    const int tid  = threadIdx.x;
    const int row0 = blockIdx.x * 16;

    for (int i = tid; i < 16 * HID; i += NTHREADS) {
        int r = i >> 7, c = i & 127;
        sX[i] = X[(size_t)(row0 + r) * HID + c];
    }

    const int lane = tid & 31;
    const int m    = lane & 15;
    const int kh   = lane >> 4;                    // 0: K+{0,1}, 1: K+{2,3}
    const int col0 = (tid >> 5) * 16;

    v8f acc = {};
    for (int k0 = 0; k0 < HID; k0 += KCHUNK) {
        __syncthreads();
        for (int i = tid; i < KCHUNK * COLS; i += NTHREADS)
            sW[i] = W[(size_t)(k0 + i / COLS) * COLS + (i % COLS)];
        __syncthreads();
#pragma unroll
        for (int k = 0; k < KCHUNK; k += 4) {
            const int ka = k + 2 * kh;
            v2f a, b;
            a.x = sX[m * HID + k0 + ka];
            a.y = sX[m * HID + k0 + ka + 1];
            b.x = sW[ka * COLS + col0 + m];
            b.y = sW[(ka + 1) * COLS + col0 + m];
            acc = __builtin_amdgcn_wmma_f32_16x16x4_f32(
                false, a, false, b, (short)0, acc, false, false);
        }
    }
#pragma unroll
    for (int r = 0; r < 8; ++r)
        Y[(size_t)(row0 + r + 8 * kh) * COLS + col0 + m] = acc[r];
}

// ---------------- self-loop init (+bias) ----------------
template <int LOGC>
__global__ void k_selfinit(const float* __restrict__ feat, const float* __restrict__ dinv,
                           const float* __restrict__ bias, float* __restrict__ out,
                           long long total) {
    long long i = (long long)blockIdx.x * blockDim.x + threadIdx.x;
    if (i >= total) return;
    long long node = i >> LOGC;
    int c = (int)(i & ((1 << LOGC) - 1));
    float di = dinv[node];
    out[i] = feat[i] * di * di + bias[c];
}

// ---------------- edge scatter: out[dst] += feat[src] * dinv[s]*dinv[d] ----------------
template <int LOGC>
__global__ void k_scatter(const long long* __restrict__ ei, int E,
                          const float* __restrict__ dinv,
                          const float* __restrict__ feat,
                          float* __restrict__ out) {
    long long idx = (long long)blockIdx.x * blockDim.x + threadIdx.x;
    long long e = idx >> LOGC;
    if (e >= E) return;
    int c = threadIdx.x & ((1 << LOGC) - 1);
    long long s = ei[e];
    long long d = ei[(long long)E + e];
    float norm = dinv[s] * dinv[d];
    unsafeAtomicAdd(out + ((size_t)d << LOGC) + c,
                    feat[((size_t)s << LOGC) + c] * norm);
}

__global__ void k_relu(float* h, long long n) {
    long long i = (long long)blockIdx.x * blockDim.x + threadIdx.x;
    if (i < n) h[i] = fmaxf(h[i], 0.0f);
}

// ---------------- host ----------------
extern "C" void kernel_launch(void* const* d_in, const int* in_sizes, int n_in,
                              void* d_out, int out_size, void* d_ws, size_t ws_size,
                              hipStream_t stream) {
    const float*     x  = (const float*)d_in[0];
    const long long* ei = (const long long*)d_in[1];
    const float*     W1 = (const float*)d_in[2];
    const float*     b1 = (const float*)d_in[3];
    const float*     W2 = (const float*)d_in[4];
    const float*     b2 = (const float*)d_in[5];
    float*           out = (float*)d_out;

    const int N = in_sizes[0] / HID;   // 50000
    const int E = in_sizes[1] / 2;     // 600000
    const int OUTC = in_sizes[5];      // 64

    // workspace layout
    float* dinv = (float*)d_ws;                            // N
    float* xw   = dinv + (((size_t)N + 511) & ~(size_t)511); // N*128
    float* agg1 = xw + (size_t)N * HID;                    // N*128 (becomes h)
    float* hw   = xw;                                      // reuse xw for h@W2 (N*64)

    const int TB = 256;
    // 1) degrees -> dinv
    k_deg_init <<<(N + TB - 1) / TB, TB, 0, stream>>>(dinv, N);
    k_deg_count<<<(E + TB - 1) / TB, TB, 0, stream>>>(ei + E, dinv, E);
    k_deg_rsqrt<<<(N + TB - 1) / TB, TB, 0, stream>>>(dinv, N);

    // 2) xw = x @ W1   (WMMA, 16 rows per block, 8 waves = 8 col tiles)
    k_gemm<HID><<<N / 16, (HID / 16) * 32, 0, stream>>>(x, W1, xw);

    // 3) agg1 = self-loop + b1, then edge scatter, then ReLU -> h (in place)
    {
        long long tot = (long long)N * HID;
        k_selfinit<7><<<(unsigned)((tot + TB - 1) / TB), TB, 0, stream>>>(xw, dinv, b1, agg1, tot);
        long long work = (long long)E * HID;
        k_scatter<7><<<(unsigned)((work + TB - 1) / TB), TB, 0, stream>>>(ei, E, dinv, xw, agg1);
        k_relu<<<(unsigned)((tot + TB - 1) / TB), TB, 0, stream>>>(agg1, tot);
    }

    // 4) hw = h @ W2   (WMMA, 4 waves = 4 col tiles of 64)
    k_gemm<64><<<N / 16, (64 / 16) * 32, 0, stream>>>(agg1, W2, hw);

    // 5) out = self-loop + b2, then edge scatter (directly into d_out)
    {
        long long tot = (long long)N * OUTC;
        k_selfinit<6><<<(unsigned)((tot + TB - 1) / TB), TB, 0, stream>>>(hw, dinv, b2, out, tot);
        long long work = (long long)E * OUTC;
        k_scatter<6><<<(unsigned)((work + TB - 1) / TB), TB, 0, stream>>>(ei, E, dinv, hw, out);
    }
}